// GraphSAGE_50062138802388
// MI455X (gfx1250) — compile-verified
//
#include <hip/hip_runtime.h>
#include <hip/hip_bf16.h>

typedef __attribute__((ext_vector_type(2))) float v2f;
typedef __attribute__((ext_vector_type(8))) float v8f;

// ---------------------------------------------------------------------------
// zero fill
// ---------------------------------------------------------------------------
__global__ void zero_f32(float* __restrict__ p, long long n) {
    long long i = (long long)blockIdx.x * blockDim.x + threadIdx.x;
    if (i < n) p[i] = 0.0f;
}

// ---------------------------------------------------------------------------
// in-degree via fp32 atomics (clamped to >=1 at use site)
// ---------------------------------------------------------------------------
__global__ void degree_kernel(const int* __restrict__ dst, float* __restrict__ deg, int nE) {
    int e = blockIdx.x * blockDim.x + threadIdx.x;
    if (e < nE) atomicAdd(&deg[dst[e]], 1.0f);
}

// ---------------------------------------------------------------------------
// edge scatter: agg[dst[e]] += h[src[e]]  (DIM floats per edge, float4 gather)
// blockDim.x == 256, DIM/4 threads per edge
// ---------------------------------------------------------------------------
template <int DIM>
__global__ __launch_bounds__(256) void scatter_add(const float* __restrict__ h,
                                                   const int* __restrict__ src,
                                                   const int* __restrict__ dst,
                                                   float* __restrict__ agg, int nE) {
    constexpr int TPE = DIM / 4;           // threads per edge
    constexpr int EPB = 256 / TPE;         // edges per block
    const int t = threadIdx.x;
    const int e = blockIdx.x * EPB + t / TPE;
    if (e >= nE) return;
    const int f = (t % TPE) * 4;
    const int s = src[e];
    const int d = dst[e];
    const float4 v = *reinterpret_cast<const float4*>(h + (size_t)s * DIM + f);
    float* o = agg + (size_t)d * DIM + f;
    atomicAdd(o + 0, v.x);
    atomicAdd(o + 1, v.y);
    atomicAdd(o + 2, v.z);
    atomicAdd(o + 3, v.w);
}

// ---------------------------------------------------------------------------
// Fused SAGE layer: out = relu(h @ Ws + (agg/deg) @ Wn + b)
//   M tiled by 16 rows per block (grid.x = M/16), NCOL/16 waves per block,
//   each wave owns one 16x16 output tile, K-loop in steps of 4 using
//   V_WMMA_F32_16X16X4_F32 (full fp32 precision).
//
// A-fragment layout (16x4 fp32): lane (l<16): M=l, {K=k0,k0+1};
//                                lane (l>=16): M=l-16, {K=k0+2,k0+3}.
// B-fragment layout (4x16 fp32): VGPR0: K=k0 (lanes 0-15) / K=k0+2 (16-31);
//                                VGPR1: K=k0+1 / K=k0+3; N = lane%16.
// C/D layout: VGPR v -> row (v + 8*half), col = lane%16.
// ---------------------------------------------------------------------------
template <int K, int NCOL>
__global__ __launch_bounds__(32 * (NCOL / 16)) void sage_layer_wmma(
    const float* __restrict__ h, const float* __restrict__ agg,
    const float* __restrict__ deg, const float* __restrict__ Ws,
    const float* __restrict__ Wn, const float* __restrict__ bias,
    float* __restrict__ out) {
    constexpr int NW = NCOL / 16;   // waves per block
    constexpr int LS = K + 4;       // padded LDS row stride (bank-conflict free)
    __shared__ float sH[16 * LS];
    __shared__ float sA[16 * LS];

    const int t = threadIdx.x;
    const int row0 = blockIdx.x * 16;

    // Cooperative, coalesced staging of the 16xK tiles of h and agg/deg.
    for (int i = t; i < 16 * K; i += 32 * NW) {
        const int r = i / K;
        const int c = i - r * K;
        const float invd = 1.0f / fmaxf(deg[row0 + r], 1.0f);
        sH[r * LS + c] = h[(size_t)(row0 + r) * K + c];
        sA[r * LS + c] = agg[(size_t)(row0 + r) * K + c] * invd;
    }
    __syncthreads();   // also guarantees all global agg reads precede stores
                       // (layer 1 writes its output in-place over agg)

    const int wave = t >> 5;
    const int lane = t & 31;
    const int half = lane >> 4;     // 0: K pair {0,1} / rows 0-7 of D
    const int lm = lane & 15;       // A row / B,C,D column within tile
    const int col0 = wave * 16;

    v8f acc = {};

    // GEMM 1: h @ Ws
#pragma unroll
    for (int k0 = 0; k0 < K; k0 += 4) {
        v2f a, b;
        a.x = sH[lm * LS + k0 + 2 * half + 0];
        a.y = sH[lm * LS + k0 + 2 * half + 1];
        b.x = Ws[(size_t)(k0 + 2 * half + 0) * NCOL + col0 + lm];
        b.y = Ws[(size_t)(k0 + 2 * half + 1) * NCOL + col0 + lm];
        acc = __builtin_amdgcn_wmma_f32_16x16x4_f32(false, a, false, b,
                                                    (short)0, acc, false, false);
    }
    // GEMM 2: (agg/deg) @ Wn
#pragma unroll
    for (int k0 = 0; k0 < K; k0 += 4) {
        v2f a, b;
        a.x = sA[lm * LS + k0 + 2 * half + 0];
        a.y = sA[lm * LS + k0 + 2 * half + 1];
        b.x = Wn[(size_t)(k0 + 2 * half + 0) * NCOL + col0 + lm];
        b.y = Wn[(size_t)(k0 + 2 * half + 1) * NCOL + col0 + lm];
        acc = __builtin_amdgcn_wmma_f32_16x16x4_f32(false, a, false, b,
                                                    (short)0, acc, false, false);
    }

    const float bc = bias[col0 + lm];
#pragma unroll
    for (int v = 0; v < 8; ++v) {
        float r = acc[v] + bc;
        r = fmaxf(r, 0.0f);
        out[(size_t)(row0 + v + 8 * half) * NCOL + col0 + lm] = r;
    }
}

// ---------------------------------------------------------------------------
// Orchestration
// ---------------------------------------------------------------------------
extern "C" void kernel_launch(void* const* d_in, const int* in_sizes, int n_in,
                              void* d_out, int out_size, void* d_ws, size_t ws_size,
                              hipStream_t stream) {
    const float* feat = (const float*)d_in[0];
    const int*   src  = (const int*)d_in[1];
    const int*   dst  = (const int*)d_in[2];
    const float* ws0 = (const float*)d_in[3];
    const float* wn0 = (const float*)d_in[4];
    const float* b0  = (const float*)d_in[5];
    const float* ws1 = (const float*)d_in[6];
    const float* wn1 = (const float*)d_in[7];
    const float* b1  = (const float*)d_in[8];
    const float* ws2 = (const float*)d_in[9];
    const float* wn2 = (const float*)d_in[10];
    const float* b2  = (const float*)d_in[11];

    const int N = in_sizes[0] / 64;   // 100000 (divisible by 16)
    const int E = in_sizes[1];        // 1600000

    float* deg = (float*)d_ws;
    float* B0  = deg + N;
    float* B1  = B0 + (size_t)N * 128;

    const int rowTiles = N / 16;

    // degrees + zeroed 64-wide accumulator for layer 0
    {
        long long n = (long long)N + (long long)N * 64;   // deg + B0[0..N*64)
        zero_f32<<<(int)((n + 255) / 256), 256, 0, stream>>>(deg, n);
    }
    degree_kernel<<<(E + 255) / 256, 256, 0, stream>>>(dst, deg, E);

    // ----- layer 0: feat(64) -> B1(128) -----
    scatter_add<64><<<(E + 15) / 16, 256, 0, stream>>>(feat, src, dst, B0, E);
    sage_layer_wmma<64, 128><<<rowTiles, 256, 0, stream>>>(feat, B0, deg,
                                                           ws0, wn0, b0, B1);

    // ----- layer 1: B1(128) -> B0(128) (output in-place over agg buffer) -----
    {
        long long n = (long long)N * 128;
        zero_f32<<<(int)((n + 255) / 256), 256, 0, stream>>>(B0, n);
    }
    scatter_add<128><<<(E + 7) / 8, 256, 0, stream>>>(B1, src, dst, B0, E);
    sage_layer_wmma<128, 128><<<rowTiles, 256, 0, stream>>>(B1, B0, deg,
                                                            ws1, wn1, b1, B0);

    // ----- layer 2: B0(128) -> d_out(64) -----
    {
        long long n = (long long)N * 128;
        zero_f32<<<(int)((n + 255) / 256), 256, 0, stream>>>(B1, n);
    }
    scatter_add<128><<<(E + 7) / 8, 256, 0, stream>>>(B0, src, dst, B1, E);
    sage_layer_wmma<128, 64><<<rowTiles, 128, 0, stream>>>(B0, B1, deg,
                                                           ws2, wn2, b2,
                                                           (float*)d_out);
}